// GraphTransformerLayer_52939766890940
// MI455X (gfx1250) — compile-verified
//
#include <hip/hip_runtime.h>
#include <hip/hip_bf16.h>
#include <cstdint>

// ---------------------------------------------------------------------------
// Graph-Transformer layer for MI455X (gfx1250, wave32).
// fp32 WMMA (v_wmma_f32_16x16x4_f32) for all dense GEMMs. Layer is L2/HBM
// bound (~10 GF vs ~60MB HBM / sub-GB L2 traffic), so fp32 is kept.
// GEMM operand panels are staged into LDS with CDNA5 async global->LDS
// copies (global_load_async_to_lds_b128, ASYNCcnt / s_wait_asynccnt) to cut
// L2 traffic ~5x vs. per-wave streaming. Edge-space GEMM (160000x512@512x64)
// is algebraically folded into node-space per-head P = V_h @ Wo_h.
// ---------------------------------------------------------------------------

typedef float v2f __attribute__((ext_vector_type(2)));
typedef float v8f __attribute__((ext_vector_type(8)));

#define HH   8
#define DD   64
#define HD   512      // H*D
#define OUTD 64
#define EPSF 1e-5f

__device__ __forceinline__ v8f wmma_f32_k4(v2f a, v2f b, v8f c) {
  // D = A(16x4) * B(4x16) + C(16x16), fp32, wave32
  return __builtin_amdgcn_wmma_f32_16x16x4_f32(
      /*neg_a=*/false, a, /*neg_b=*/false, b,
      /*c_mod=*/(short)0, c, /*reuse_a=*/false, /*reuse_b=*/false);
}

// CDNA5 async global->LDS copy, 16B per lane, tracked by ASYNCcnt.
// vdst = per-lane LDS byte address, vaddr = per-lane 64-bit global address.
__device__ __forceinline__ void async_copy_f4(uint32_t lds_off, const float* g) {
  asm volatile("global_load_async_to_lds_b128 %0, %1, off"
               :: "v"(lds_off), "v"(g)
               : "memory");
}
__device__ __forceinline__ void wait_async0() {
  asm volatile("s_wait_asynccnt 0x0" ::: "memory");
}
__device__ __forceinline__ uint32_t lds_addr(const void* p) {
  // generic shared pointer = {aperture_hi32, lds_offset_lo32}
  return (uint32_t)(uintptr_t)p;
}

// --------------------------- 0: zero scratch -------------------------------
__global__ __launch_bounds__(256) void k_zero(float* p, int n) {
  int i = blockIdx.x * blockDim.x + threadIdx.x;
  if (i < n) p[i] = 0.0f;
}

// --------------------- 1: QKV projection (WMMA GEMM) -----------------------
// Y = X[M,512] @ W[512,512]. Workgroup (8 waves) computes 128(M)x64(N);
// B panel staged 32x64 at a time via async-to-LDS; A streamed from L2.
#define KC 32
__global__ __launch_bounds__(256)
void k_gemm_qkv(const float* __restrict__ X,
                const float* __restrict__ W0, const float* __restrict__ W1,
                const float* __restrict__ W2,
                float* __restrict__ Y0, float* __restrict__ Y1,
                float* __restrict__ Y2, int M) {
  __shared__ float shB[KC * 64];              // 8 KB panel
  const float* W;
  float* Y;
  if (blockIdx.z == 0)      { W = W0; Y = Y0; }
  else if (blockIdx.z == 1) { W = W1; Y = Y1; }
  else                      { W = W2; Y = Y2; }

  int t = threadIdx.x;
  int w = t >> 5, lane = t & 31;
  int rowBase = blockIdx.x * 128 + w * 16;    // 16 rows per wave
  int colBlk  = blockIdx.y * 64;              // 64 cols per block
  bool act = (rowBase + 16) <= M;
  int row   = rowBase + (lane & 15);
  int khalf = (lane >> 4) * 2;

  int sr0 = t >> 4;                           // staging: row within pass
  int sc0 = (t & 15) * 4;                     // staging: col (float4 granule)

  v8f acc[4] = {};
  for (int kc = 0; kc < HD; kc += KC) {
    // stage B[kc..kc+31, colBlk..colBlk+63] -> LDS (2 passes x 256 lanes x 16B)
    for (int p = 0; p < KC / 16; ++p) {
      int r = p * 16 + sr0;
      async_copy_f4(lds_addr(&shB[r * 64 + sc0]),
                    W + (size_t)(kc + r) * HD + colBlk + sc0);
    }
    wait_async0();
    __syncthreads();
    if (act) {
      for (int kk = 0; kk < KC; kk += 4) {
        v2f a;
        a.x = X[(size_t)row * HD + kc + kk + khalf];
        a.y = X[(size_t)row * HD + kc + kk + khalf + 1];
        for (int ct = 0; ct < 4; ++ct) {
          v2f b;
          b.x = shB[(kk + khalf) * 64 + ct * 16 + (lane & 15)];
          b.y = shB[(kk + khalf + 1) * 64 + ct * 16 + (lane & 15)];
          acc[ct] = wmma_f32_k4(a, b, acc[ct]);
        }
      }
    }
    __syncthreads();
  }
  if (act) {
    int rbase = rowBase + 8 * (lane >> 4);    // C/D: M = v + 8*(lane>>4)
    for (int ct = 0; ct < 4; ++ct) {
      int col = colBlk + ct * 16 + (lane & 15);
      for (int v = 0; v < 8; ++v)
        Y[(size_t)(rbase + v) * HD + col] = acc[ct][v];
    }
  }
}

// ------ 2: E = edge_attr @ We fused with per-head squared-norm reduce ------
// E2[e,h] = sum_d (edge_attr[e] @ We)[h*64+d]^2 — E never materialized.
// Workgroup covers 128 edges; We head-slices (64x64) staged via async-to-LDS
// and shared by all 8 waves; A (edge rows) cached in registers.
__global__ __launch_bounds__(256)
void k_e_norm(const float* __restrict__ EA,   // [NE,64]
              const float* __restrict__ We,   // [64,512]
              float* __restrict__ E2,         // [NE,8]
              int NE) {
  __shared__ float shB[64 * 64];              // 16 KB head slice
  int t = threadIdx.x;
  int w = t >> 5, lane = t & 31;
  int te  = blockIdx.x * 8 + w;               // 16-edge tile per wave
  bool act = te * 16 < NE;
  int row   = te * 16 + (lane & 15);
  int khalf = (lane >> 4) * 2;

  v2f a[16];                                  // full K=64 A fragments
  if (act) {
    for (int ks = 0; ks < 16; ++ks) {
      a[ks].x = EA[(size_t)row * 64 + ks * 4 + khalf];
      a[ks].y = EA[(size_t)row * 64 + ks * 4 + khalf + 1];
    }
  }

  int sr0 = t >> 4, sc0 = (t & 15) * 4;
  for (int h = 0; h < HH; ++h) {
    // stage We[:, h*64 .. h*64+63] (64x64) -> LDS, 4 passes
    for (int p = 0; p < 4; ++p) {
      int r = p * 16 + sr0;
      async_copy_f4(lds_addr(&shB[r * 64 + sc0]),
                    We + (size_t)r * HD + h * DD + sc0);
    }
    wait_async0();
    __syncthreads();
    if (act) {
      float part[8];
      for (int v = 0; v < 8; ++v) part[v] = 0.0f;
      for (int ct = 0; ct < 4; ++ct) {        // 4 col tiles per head
        v8f c = {};
        for (int ks = 0; ks < 16; ++ks) {
          v2f b;
          b.x = shB[(ks * 4 + khalf) * 64 + ct * 16 + (lane & 15)];
          b.y = shB[(ks * 4 + khalf + 1) * 64 + ct * 16 + (lane & 15)];
          c = wmma_f32_k4(a[ks], b, c);
        }
        for (int v = 0; v < 8; ++v) part[v] += c[v] * c[v];
      }
      for (int m = 1; m < 16; m <<= 1)        // row sum over 16 lanes/half
        for (int v = 0; v < 8; ++v) part[v] += __shfl_xor(part[v], m, 32);
      if ((lane & 15) == 0) {
        int rbase = te * 16 + 8 * (lane >> 4);
        for (int v = 0; v < 8; ++v) E2[(rbase + v) * HH + h] = part[v];
      }
    }
    __syncthreads();
  }
}

// ------------- 3: per-head projection P[n,h,:] = V[n,h,:] @ Wo_h -----------
__global__ __launch_bounds__(256)
void k_pproj(const float* __restrict__ V,     // [N,512]
             const float* __restrict__ Wo,    // [512,64]
             float* __restrict__ P,           // [N,512] (h-major blocks)
             int M) {
  int wave = (blockIdx.x * blockDim.x + threadIdx.x) >> 5;
  int lane = threadIdx.x & 31;
  int ct = wave & 3;
  int h  = (wave >> 2) & 7;
  int tm = wave >> 5;
  if (tm * 16 >= M) return;

  int row   = tm * 16 + (lane & 15);
  int col   = ct * 16 + (lane & 15);
  int khalf = (lane >> 4) * 2;

  v8f c = {};
  for (int k = 0; k < DD; k += 4) {
    v2f a, b;
    a.x = V[(size_t)row * HD + h * DD + k + khalf];
    a.y = V[(size_t)row * HD + h * DD + k + khalf + 1];
    b.x = Wo[(h * DD + k + khalf) * OUTD + col];
    b.y = Wo[(h * DD + k + khalf + 1) * OUTD + col];
    c = wmma_f32_k4(a, b, c);
  }
  int rbase = tm * 16 + 8 * (lane >> 4);
  for (int v = 0; v < 8; ++v) P[(size_t)(rbase + v) * HD + h * DD + col] = c[v];
}

// --------- 4: per-edge scores + softmax over heads (wave per edge) ---------
__global__ __launch_bounds__(256)
void k_scores(const float* __restrict__ Q, const float* __restrict__ K,
              const float* __restrict__ E2,
              const int* __restrict__ src, const int* __restrict__ dst,
              float* __restrict__ S, int NE) {
  int e    = (blockIdx.x * blockDim.x + threadIdx.x) >> 5;
  int lane = threadIdx.x & 31;
  if (e >= NE) return;
  int sn = src[e], dn = dst[e];

  const float4* q = (const float4*)(Q + (size_t)sn * HD);
  const float4* k = (const float4*)(K + (size_t)dn * HD);
  float acc = 0.0f;
  int b4 = lane * 4;                          // lane covers dims lane*16..+15
  for (int j = 0; j < 4; ++j) {
    float4 qv = q[b4 + j], kv = k[b4 + j];
    acc += qv.x * kv.x + qv.y * kv.y + qv.z * kv.z + qv.w * kv.w;
  }
  acc += __shfl_xor(acc, 1, 32);
  acc += __shfl_xor(acc, 2, 32);
  int h = lane >> 2;
  float val = (acc + E2[e * HH + h]) * 0.125f;   // /sqrt(64)

  float mx = val;
  mx = fmaxf(mx, __shfl_xor(mx, 4, 32));
  mx = fmaxf(mx, __shfl_xor(mx, 8, 32));
  mx = fmaxf(mx, __shfl_xor(mx, 16, 32));
  float ex  = expf(val - mx);
  float sum = ex;
  sum += __shfl_xor(sum, 4, 32);
  sum += __shfl_xor(sum, 8, 32);
  sum += __shfl_xor(sum, 16, 32);
  if ((lane & 3) == 0) S[e * HH + h] = ex / sum;
}

// ------- 5: per-edge combine y = sum_h s[e,h]*P[dst,h,:], scatter-add ------
__global__ __launch_bounds__(256)
void k_aggregate(const float* __restrict__ P, const float* __restrict__ S,
                 const int* __restrict__ src, const int* __restrict__ dst,
                 float* __restrict__ sums, float* __restrict__ cnt, int NE) {
  int e    = (blockIdx.x * blockDim.x + threadIdx.x) >> 5;
  int lane = threadIdx.x & 31;
  if (e >= NE) return;
  int sn = src[e], dn = dst[e];

  float sc[HH];
  for (int h = 0; h < HH; ++h) sc[h] = S[e * HH + h];
  const float* p = P + (size_t)dn * HD;
  float y0 = 0.0f, y1 = 0.0f;
  for (int h = 0; h < HH; ++h) {
    y0 += sc[h] * p[h * DD + lane];
    y1 += sc[h] * p[h * DD + 32 + lane];
  }
  atomicAdd(&sums[sn * OUTD + lane], y0);
  atomicAdd(&sums[sn * OUTD + 32 + lane], y1);
  if (lane == 0) atomicAdd(&cnt[sn], 1.0f);
}

// ---- 6: scatter-mean + LayerNorm1 (wave per node) + BN1 stat accumulate ---
__global__ __launch_bounds__(256)
void k_mean_ln1(const float* __restrict__ sums, const float* __restrict__ cnt,
                const float* __restrict__ g, const float* __restrict__ b,
                float* __restrict__ h1,
                float* __restrict__ colsum, float* __restrict__ colsq, int Nn) {
  __shared__ float ssum[OUTD], ssq[OUTD];
  int t = threadIdx.x;
  if (t < OUTD) { ssum[t] = 0.0f; ssq[t] = 0.0f; }
  __syncthreads();

  int node = (blockIdx.x * blockDim.x + t) >> 5;
  int lane = t & 31;
  if (node < Nn) {
    float c  = fmaxf(cnt[node], 1.0f);
    float v0 = sums[node * OUTD + lane] / c;
    float v1 = sums[node * OUTD + 32 + lane] / c;
    float m = v0 + v1;
    for (int msk = 1; msk < 32; msk <<= 1) m += __shfl_xor(m, msk, 32);
    m *= (1.0f / 64.0f);
    float d0 = v0 - m, d1 = v1 - m;
    float var = d0 * d0 + d1 * d1;
    for (int msk = 1; msk < 32; msk <<= 1) var += __shfl_xor(var, msk, 32);
    var *= (1.0f / 64.0f);
    float rs = rsqrtf(var + EPSF);
    v0 = d0 * rs * g[lane] + b[lane];
    v1 = d1 * rs * g[lane + 32] + b[lane + 32];
    h1[node * OUTD + lane] = v0;
    h1[node * OUTD + 32 + lane] = v1;
    atomicAdd(&ssum[lane], v0);       atomicAdd(&ssum[lane + 32], v1);
    atomicAdd(&ssq[lane], v0 * v0);   atomicAdd(&ssq[lane + 32], v1 * v1);
  }
  __syncthreads();
  if (t < OUTD) { atomicAdd(&colsum[t], ssum[t]); atomicAdd(&colsq[t], ssq[t]); }
}

// ----------------------- 7/9: BatchNorm stat finalize ----------------------
__global__ void k_bn_finalize(const float* __restrict__ colsum,
                              const float* __restrict__ colsq,
                              float* __restrict__ mean, float* __restrict__ invstd,
                              float invN) {
  int c = threadIdx.x;
  float m = colsum[c] * invN;
  float v = colsq[c] * invN - m * m;
  mean[c] = m;
  invstd[c] = rsqrtf(v + EPSF);
}

// --- 8: BN1 apply + MLP(64->128->64) + LayerNorm2 + BN2 stat accumulate ----
__global__ __launch_bounds__(256)
void k_mlp_ln2(const float* __restrict__ h1,
               const float* __restrict__ mean1, const float* __restrict__ is1,
               const float* __restrict__ bg, const float* __restrict__ bb,
               const float* __restrict__ W1, const float* __restrict__ b1,
               const float* __restrict__ W2, const float* __restrict__ b2,
               const float* __restrict__ lg, const float* __restrict__ lb,
               float* __restrict__ h2,
               float* __restrict__ colsum, float* __restrict__ colsq, int Nn) {
  __shared__ float Z[8][OUTD];
  __shared__ float T[8][2 * OUTD];
  __shared__ float ssum[OUTD], ssq[OUTD];
  int t = threadIdx.x;
  if (t < OUTD) { ssum[t] = 0.0f; ssq[t] = 0.0f; }
  int w = t >> 5, lane = t & 31;
  int node = (blockIdx.x << 3) + w;
  bool act = node < Nn;

  if (act) {
    float z0 = (h1[node * OUTD + lane] - mean1[lane]) * is1[lane] * bg[lane] + bb[lane];
    float z1 = (h1[node * OUTD + 32 + lane] - mean1[lane + 32]) * is1[lane + 32] * bg[lane + 32] + bb[lane + 32];
    Z[w][lane] = z0; Z[w][lane + 32] = z1;
  }
  __syncthreads();
  if (act) {
    for (int r = 0; r < 4; ++r) {
      int j = lane + 32 * r;
      float accv = b1[j];
      for (int cdim = 0; cdim < OUTD; ++cdim) accv += Z[w][cdim] * W1[cdim * 128 + j];
      T[w][j] = fmaxf(accv, 0.0f);
    }
  }
  __syncthreads();
  if (act) {
    float o0 = b2[lane], o1 = b2[lane + 32];
    for (int j = 0; j < 128; ++j) {
      float tv = T[w][j];
      o0 += tv * W2[j * OUTD + lane];
      o1 += tv * W2[j * OUTD + lane + 32];
    }
    float m = o0 + o1;
    for (int msk = 1; msk < 32; msk <<= 1) m += __shfl_xor(m, msk, 32);
    m *= (1.0f / 64.0f);
    float d0 = o0 - m, d1 = o1 - m;
    float var = d0 * d0 + d1 * d1;
    for (int msk = 1; msk < 32; msk <<= 1) var += __shfl_xor(var, msk, 32);
    var *= (1.0f / 64.0f);
    float rs = rsqrtf(var + EPSF);
    o0 = d0 * rs * lg[lane] + lb[lane];
    o1 = d1 * rs * lg[lane + 32] + lb[lane + 32];
    h2[node * OUTD + lane] = o0;
    h2[node * OUTD + 32 + lane] = o1;
    atomicAdd(&ssum[lane], o0);       atomicAdd(&ssum[lane + 32], o1);
    atomicAdd(&ssq[lane], o0 * o0);   atomicAdd(&ssq[lane + 32], o1 * o1);
  }
  __syncthreads();
  if (t < OUTD) { atomicAdd(&colsum[t], ssum[t]); atomicAdd(&colsq[t], ssq[t]); }
}

// ------------------------- 10: BN2 apply -> output -------------------------
__global__ __launch_bounds__(256)
void k_bn2_apply(const float* __restrict__ h2,
                 const float* __restrict__ mean, const float* __restrict__ invstd,
                 const float* __restrict__ g, const float* __restrict__ b,
                 float* __restrict__ out, int total) {
  int i = blockIdx.x * blockDim.x + threadIdx.x;
  if (i < total) {
    int c = i & 63;
    out[i] = (h2[i] - mean[c]) * invstd[c] * g[c] + b[c];
  }
}

// ---------------------------------------------------------------------------
extern "C" void kernel_launch(void* const* d_in, const int* in_sizes, int n_in,
                              void* d_out, int out_size, void* d_ws, size_t ws_size,
                              hipStream_t stream) {
  const float* x    = (const float*)d_in[0];
  const int*   eidx = (const int*)  d_in[1];
  const float* ea   = (const float*)d_in[2];
  const float* Wq   = (const float*)d_in[3];
  const float* Wk   = (const float*)d_in[4];
  const float* Wv   = (const float*)d_in[5];
  const float* We   = (const float*)d_in[6];
  const float* Wo   = (const float*)d_in[7];
  const float* ln1g = (const float*)d_in[8];
  const float* ln1b = (const float*)d_in[9];
  const float* bn1g = (const float*)d_in[10];
  const float* bn1b = (const float*)d_in[11];
  const float* W1   = (const float*)d_in[12];
  const float* b1   = (const float*)d_in[13];
  const float* W2   = (const float*)d_in[14];
  const float* b2   = (const float*)d_in[15];
  const float* ln2g = (const float*)d_in[16];
  const float* ln2b = (const float*)d_in[17];
  const float* bn2g = (const float*)d_in[18];
  const float* bn2b = (const float*)d_in[19];

  const int Nn = in_sizes[0] / HD;        // 10000
  const int NE = in_sizes[1] / 2;         // 160000
  const int* src = eidx;
  const int* dst = eidx + NE;

  // -------- workspace layout (floats) --------
  float* ws = (float*)d_ws;
  size_t off = 0;
  float* Q    = ws + off; off += (size_t)Nn * HD;
  float* K    = ws + off; off += (size_t)Nn * HD;
  float* V    = ws + off; off += (size_t)Nn * HD;
  float* P    = ws + off; off += (size_t)Nn * HD;
  float* E2   = ws + off; off += (size_t)NE * HH;
  float* S    = ws + off; off += (size_t)NE * HH;
  float* sums = ws + off; off += (size_t)Nn * OUTD;
  float* cnt  = ws + off; off += (size_t)Nn;
  float* st   = ws + off; off += 512;     // BN accumulators + finalized stats
  float* h1   = ws + off; off += (size_t)Nn * OUTD;
  float* h2   = ws + off; off += (size_t)Nn * OUTD;
  float* cs1 = st, *cq1 = st + 64, *cs2 = st + 128, *cq2 = st + 192;
  float* mu1 = st + 256, *is1 = st + 320, *mu2 = st + 384, *is2 = st + 448;

  // 0) zero sums/cnt/BN accumulators (contiguous span)
  int zn = Nn * OUTD + Nn + 512;
  k_zero<<<(zn + 255) / 256, 256, 0, stream>>>(sums, zn);

  // 1) Q,K,V projections (WMMA, LDS-tiled, async B staging)
  dim3 gq((Nn + 127) / 128, HD / 64, 3);
  k_gemm_qkv<<<gq, 256, 0, stream>>>(x, Wq, Wk, Wv, Q, K, V, Nn);

  // 2) E projection fused with per-head squared norm (WMMA, async We staging)
  k_e_norm<<<(NE + 127) / 128, 256, 0, stream>>>(ea, We, E2, NE);

  // 3) per-head P = V_h @ Wo_h (WMMA)
  int tilesM = (Nn + 15) / 16;
  k_pproj<<<(tilesM * 32 * 32 + 255) / 256, 256, 0, stream>>>(V, Wo, P, Nn);

  // 4) per-edge scores + head softmax
  k_scores<<<((size_t)NE * 32 + 255) / 256, 256, 0, stream>>>(Q, K, E2, src, dst, S, NE);

  // 5) per-edge combine + scatter-add
  k_aggregate<<<((size_t)NE * 32 + 255) / 256, 256, 0, stream>>>(P, S, src, dst, sums, cnt, NE);

  // 6) mean + LN1 + BN1 stats
  k_mean_ln1<<<((size_t)Nn * 32 + 255) / 256, 256, 0, stream>>>(sums, cnt, ln1g, ln1b, h1, cs1, cq1, Nn);

  // 7) BN1 finalize
  k_bn_finalize<<<1, 64, 0, stream>>>(cs1, cq1, mu1, is1, 1.0f / (float)Nn);

  // 8) BN1 apply + MLP + LN2 + BN2 stats
  k_mlp_ln2<<<(Nn + 7) / 8, 256, 0, stream>>>(h1, mu1, is1, bn1g, bn1b,
                                              W1, b1, W2, b2, ln2g, ln2b,
                                              h2, cs2, cq2, Nn);

  // 9) BN2 finalize
  k_bn_finalize<<<1, 64, 0, stream>>>(cs2, cq2, mu2, is2, 1.0f / (float)Nn);

  // 10) BN2 apply -> output
  k_bn2_apply<<<(Nn * OUTD + 255) / 256, 256, 0, stream>>>(h2, mu2, is2, bn2g, bn2b,
                                                           (float*)d_out, Nn * OUTD);
}